// GNN_3_87316685128354
// MI455X (gfx1250) — compile-verified
//
#include <hip/hip_runtime.h>

#define N_NODES  50000
#define N_EDGES  800000
#define N_GRAPHS 128
#define HDIM     256

typedef __attribute__((ext_vector_type(2))) float v2f;
typedef __attribute__((ext_vector_type(8))) float v8f;

// ---------------- degree / normalization ----------------

__global__ __launch_bounds__(256) void k_deg_init(float* __restrict__ deg) {
  int i = blockIdx.x * 256 + threadIdx.x;
  if (i < N_NODES) deg[i] = 1.0f;   // self-loop contributes 1.0
}

__global__ __launch_bounds__(256) void k_deg_acc(const int* __restrict__ dst,
                                                 const float* __restrict__ ea,
                                                 float* __restrict__ deg) {
  int e = blockIdx.x * 256 + threadIdx.x;
  if (e < N_EDGES) unsafeAtomicAdd(&deg[dst[e]], ea[e]);
}

__global__ __launch_bounds__(256) void k_dinv(float* __restrict__ deg) {
  int i = blockIdx.x * 256 + threadIdx.x;
  if (i < N_NODES) {
    float d = deg[i];
    deg[i] = (d > 0.0f) ? rsqrtf(d) : 0.0f;   // in-place: deg -> dinv
  }
}

__global__ __launch_bounds__(256) void k_norm(const int* __restrict__ src,
                                              const int* __restrict__ dst,
                                              const float* __restrict__ ea,
                                              const float* __restrict__ dinv,
                                              float* __restrict__ nrm) {
  int e = blockIdx.x * 256 + threadIdx.x;
  if (e < N_EDGES) nrm[e] = dinv[src[e]] * ea[e] * dinv[dst[e]];
}

// ---------------- W repack into WMMA-fragment-native layout ----------------
// Wp[((k8*256 + j)*2 + hi)*4 + q] = W[(8*k8 + 2*hi + 4*(q>>1) + (q&1))*256 + j]
// -> one float4 per lane per tile delivers B VGPR0/VGPR1 for TWO K-steps,
//    and a wave's 32 lanes cover a contiguous 512B span.

__global__ __launch_bounds__(256) void k_repack(const float* __restrict__ W,
                                                float* __restrict__ Wp) {
  int idx = blockIdx.x * 256 + threadIdx.x;          // over 256*256
  if (idx >= HDIM * HDIM) return;
  int k = idx >> 8, j = idx & 255;
  int k8  = k >> 3;
  int rem = k & 7;
  int hi  = (rem >> 1) & 1;
  int q   = (rem & 1) + ((rem >> 2) << 1);
  Wp[(((size_t)k8 * 256 + j) * 2 + hi) * 4 + q] = W[idx];
}

// ---------------- fp32 WMMA GEMM: out[N,256] = act(A)[N,256] @ W[256,256] ----------------
// One wave per 16x64 output tile; 4 accumulators; 32 k8-iterations x 8 wmma.
// Per iteration: 2x global_load_b64 (A) + 4x global_load_b128 (packed B) feed 8 WMMAs.

template <bool RELU_IN>
__global__ __launch_bounds__(32) void k_gemm_wmma(const float* __restrict__ A,
                                                  const float* __restrict__ Wp,
                                                  float* __restrict__ out) {
  const int lane = threadIdx.x;
  const int r  = lane & 15;    // M (A), N (B/C) index within half
  const int hi = lane >> 4;    // upper-lane half: K+2/K+3 (A,B), M+8 (C)
  const int r0 = blockIdx.x * 16;
  const int c0 = blockIdx.y * 64;

  const float*  arow = A + (size_t)(r0 + r) * HDIM + 2 * hi;
  const float4* bp   = (const float4*)Wp + ((size_t)(c0 + r) * 2 + hi);

  v8f acc0 = {}, acc1 = {}, acc2 = {}, acc3 = {};

  for (int k8 = 0; k8 < 32; ++k8) {
    v2f a0 = *(const v2f*)(arow + k8 * 8);        // K = 8k8+2hi, +1
    v2f a1 = *(const v2f*)(arow + k8 * 8 + 4);    // K = 8k8+4+2hi, +1
    if (RELU_IN) {
      a0.x = fmaxf(a0.x, 0.0f); a0.y = fmaxf(a0.y, 0.0f);
      a1.x = fmaxf(a1.x, 0.0f); a1.y = fmaxf(a1.y, 0.0f);
    }

    const float4* bk = bp + (size_t)k8 * 512;     // 2048 floats per k8
    float4 B0 = bk[0];                            // tile 0 (cols c0+ 0..15)
    float4 B1 = bk[32];                           // tile 1 (cols c0+16..31)
    float4 B2 = bk[64];                           // tile 2 (cols c0+32..47)
    float4 B3 = bk[96];                           // tile 3 (cols c0+48..63)

    v2f b;
    b.x = B0.x; b.y = B0.y;
    acc0 = __builtin_amdgcn_wmma_f32_16x16x4_f32(false, a0, false, b, (short)0, acc0, false, false);
    b.x = B1.x; b.y = B1.y;
    acc1 = __builtin_amdgcn_wmma_f32_16x16x4_f32(false, a0, false, b, (short)0, acc1, false, false);
    b.x = B2.x; b.y = B2.y;
    acc2 = __builtin_amdgcn_wmma_f32_16x16x4_f32(false, a0, false, b, (short)0, acc2, false, false);
    b.x = B3.x; b.y = B3.y;
    acc3 = __builtin_amdgcn_wmma_f32_16x16x4_f32(false, a0, false, b, (short)0, acc3, false, false);

    b.x = B0.z; b.y = B0.w;
    acc0 = __builtin_amdgcn_wmma_f32_16x16x4_f32(false, a1, false, b, (short)0, acc0, false, false);
    b.x = B1.z; b.y = B1.w;
    acc1 = __builtin_amdgcn_wmma_f32_16x16x4_f32(false, a1, false, b, (short)0, acc1, false, false);
    b.x = B2.z; b.y = B2.w;
    acc2 = __builtin_amdgcn_wmma_f32_16x16x4_f32(false, a1, false, b, (short)0, acc2, false, false);
    b.x = B3.z; b.y = B3.w;
    acc3 = __builtin_amdgcn_wmma_f32_16x16x4_f32(false, a1, false, b, (short)0, acc3, false, false);
  }

  // C/D layout: VGPR p -> row M = p + 8*hi, col N = r
  float* orow = out + (size_t)(r0 + 8 * hi) * HDIM + c0 + r;
#pragma unroll
  for (int p = 0; p < 8; ++p) {
    orow[(size_t)p * HDIM + 0]  = acc0[p];
    orow[(size_t)p * HDIM + 16] = acc1[p];
    orow[(size_t)p * HDIM + 32] = acc2[p];
    orow[(size_t)p * HDIM + 48] = acc3[p];
  }
}

// ---------------- c = self_coef * hW + bias (writes output region, pre-atomics) ----------------

__global__ __launch_bounds__(256) void k_init_c(const float* __restrict__ hW,
                                                const float* __restrict__ dinv,
                                                const float* __restrict__ bias,
                                                float* __restrict__ c) {
  int idx = blockIdx.x * 256 + threadIdx.x;            // over N*64 float4 chunks
  if (idx >= N_NODES * 64) return;
  int n = idx >> 6, q = idx & 63;
  float sc = dinv[n]; sc *= sc;                        // self_coef = dinv^2
  float4 h = ((const float4*)hW)[idx];
  float4 b = ((const float4*)bias)[q];
  float4 o;
  o.x = fmaf(sc, h.x, b.x);
  o.y = fmaf(sc, h.y, b.y);
  o.z = fmaf(sc, h.z, b.z);
  o.w = fmaf(sc, h.w, b.w);
  ((float4*)c)[idx] = o;
}

// ---------------- edge scatter: c[dst] += norm[e] * hW[src] ----------------
// 64 consecutive lanes cover one edge's 256 features -> coalesced gather + contiguous atomics.

__global__ __launch_bounds__(256) void k_edge(const int* __restrict__ src,
                                              const int* __restrict__ dst,
                                              const float* __restrict__ nrm,
                                              const float* __restrict__ hW,
                                              float* __restrict__ c) {
  int idx = blockIdx.x * 256 + threadIdx.x;            // over E*64 float4 chunks
  if (idx >= N_EDGES * 64) return;
  int e = idx >> 6, q = idx & 63;
  int s = src[e], d = dst[e];
  float w = nrm[e];
  float4 v = ((const float4*)hW)[(size_t)s * 64 + q];
  float* cp = c + (size_t)d * HDIM + q * 4;
  unsafeAtomicAdd(cp + 0, w * v.x);
  unsafeAtomicAdd(cp + 1, w * v.y);
  unsafeAtomicAdd(cp + 2, w * v.z);
  unsafeAtomicAdd(cp + 3, w * v.w);
}

// ---------------- global mean pool ----------------

__global__ __launch_bounds__(256) void k_pool_zero(float* __restrict__ psum) {
  int i = blockIdx.x * 256 + threadIdx.x;
  if (i < N_GRAPHS * HDIM + N_GRAPHS) psum[i] = 0.0f;  // psum followed by cnt
}

__global__ __launch_bounds__(256) void k_pool_acc(const float* __restrict__ c3,
                                                  const int* __restrict__ batch,
                                                  float* __restrict__ psum,
                                                  float* __restrict__ cnt) {
  int idx = blockIdx.x * 256 + threadIdx.x;            // over N*64 float4 chunks
  if (idx >= N_NODES * 64) return;
  int n = idx >> 6, q = idx & 63;
  int g = batch[n];
  float4 v = ((const float4*)c3)[idx];                 // relu(c3) on the fly
  v.x = fmaxf(v.x, 0.0f); v.y = fmaxf(v.y, 0.0f);
  v.z = fmaxf(v.z, 0.0f); v.w = fmaxf(v.w, 0.0f);
  float* p = psum + (size_t)g * HDIM + q * 4;
  unsafeAtomicAdd(p + 0, v.x);
  unsafeAtomicAdd(p + 1, v.y);
  unsafeAtomicAdd(p + 2, v.z);
  unsafeAtomicAdd(p + 3, v.w);
  if (q == 0) unsafeAtomicAdd(&cnt[g], 1.0f);
}

__global__ __launch_bounds__(256) void k_pool_fin(const float* __restrict__ psum,
                                                  const float* __restrict__ cnt,
                                                  float* __restrict__ pooled) {
  int i = blockIdx.x * 256 + threadIdx.x;
  if (i < N_GRAPHS * HDIM) pooled[i] = psum[i] / fmaxf(cnt[i >> 8], 1.0f);
}

// ---------------- tiny MLP head: 256 -> 64 -> 32 -> 2, relu everywhere ----------------

__global__ __launch_bounds__(128) void k_mlp(const float* __restrict__ pooled,
                                             const float* __restrict__ L1w, const float* __restrict__ L1b,
                                             const float* __restrict__ L2w, const float* __restrict__ L2b,
                                             const float* __restrict__ L3w, const float* __restrict__ L3b,
                                             float* __restrict__ out) {
  __shared__ float o1[128][65];   // pad to dodge bank conflicts
  __shared__ float o2[128][33];
  int g = threadIdx.x;            // one graph per lane, 128 threads

  const float* pr = pooled + (size_t)g * HDIM;
  for (int j = 0; j < 64; ++j) {
    float s = L1b[j];
    for (int k = 0; k < HDIM; ++k) s = fmaf(pr[k], L1w[k * 64 + j], s);
    o1[g][j] = fmaxf(s, 0.0f);
  }
  for (int j = 0; j < 32; ++j) {
    float s = L2b[j];
    for (int k = 0; k < 64; ++k) s = fmaf(o1[g][k], L2w[k * 32 + j], s);
    o2[g][j] = fmaxf(s, 0.0f);
  }
  for (int j = 0; j < 2; ++j) {
    float s = L3b[j];
    for (int k = 0; k < 32; ++k) s = fmaf(o2[g][k], L3w[k * 2 + j], s);
    out[g * 2 + j] = fmaxf(s, 0.0f);
  }
}

// ---------------- launcher ----------------

extern "C" void kernel_launch(void* const* d_in, const int* in_sizes, int n_in,
                              void* d_out, int out_size, void* d_ws, size_t ws_size,
                              hipStream_t stream) {
  const float* x     = (const float*)d_in[0];
  const int*   eidx  = (const int*)d_in[1];
  const int*   src   = eidx;                 // edge_index[0]
  const int*   dst   = eidx + N_EDGES;       // edge_index[1]
  const float* ea    = (const float*)d_in[2];
  const int*   batch = (const int*)d_in[3];
  const float* W1 = (const float*)d_in[4];  const float* b1 = (const float*)d_in[5];
  const float* W2 = (const float*)d_in[6];  const float* b2 = (const float*)d_in[7];
  const float* W3 = (const float*)d_in[8];  const float* b3 = (const float*)d_in[9];
  const float* L1w = (const float*)d_in[10]; const float* L1b = (const float*)d_in[11];
  const float* L2w = (const float*)d_in[12]; const float* L2b = (const float*)d_in[13];
  const float* L3w = (const float*)d_in[14]; const float* L3b = (const float*)d_in[15];

  // d_out layout = (out, pooled, c1, c2, c3) concatenated flat
  float* out    = (float*)d_out;
  float* pooled = out + (size_t)N_GRAPHS * 2;
  float* c1     = pooled + (size_t)N_GRAPHS * HDIM;
  float* c2     = c1 + (size_t)N_NODES * HDIM;
  float* c3     = c2 + (size_t)N_NODES * HDIM;

  // scratch: dinv[N] | norm[E] | hW[N*H] | Wp[256*256] | psum[128*256] | cnt[128]
  const size_t ws_need = ((size_t)N_NODES + N_EDGES + (size_t)N_NODES * HDIM +
                          (size_t)HDIM * HDIM +
                          (size_t)N_GRAPHS * HDIM + N_GRAPHS) * sizeof(float);
  if (ws_size < ws_need) return;
  float* dinv = (float*)d_ws;
  float* nrm  = dinv + N_NODES;
  float* hW   = nrm + N_EDGES;
  float* Wp   = hW + (size_t)N_NODES * HDIM;
  float* psum = Wp + (size_t)HDIM * HDIM;
  float* cnt  = psum + (size_t)N_GRAPHS * HDIM;

  const int TB = 256;
  const int gN  = (N_NODES + TB - 1) / TB;
  const int gE  = (N_EDGES + TB - 1) / TB;
  const int gNH = (N_NODES * 64 + TB - 1) / TB;   // N*H/4 float4 chunks
  const int gEH = (N_EDGES * 64 + TB - 1) / TB;   // E*H/4 float4 chunks
  const int gW  = (HDIM * HDIM + TB - 1) / TB;
  const dim3 gg(N_NODES / 16, HDIM / 64);          // 3125 x 4 waves

  // normalization
  k_deg_init<<<gN, TB, 0, stream>>>(dinv);
  k_deg_acc<<<gE, TB, 0, stream>>>(dst, ea, dinv);
  k_dinv<<<gN, TB, 0, stream>>>(dinv);
  k_norm<<<gE, TB, 0, stream>>>(src, dst, ea, dinv, nrm);

  // layer 1: A = x
  k_repack<<<gW, TB, 0, stream>>>(W1, Wp);
  k_gemm_wmma<false><<<gg, 32, 0, stream>>>(x, Wp, hW);
  k_init_c<<<gNH, TB, 0, stream>>>(hW, dinv, b1, c1);
  k_edge<<<gEH, TB, 0, stream>>>(src, dst, nrm, hW, c1);

  // layer 2: A = relu(c1), fused into the WMMA A-load
  k_repack<<<gW, TB, 0, stream>>>(W2, Wp);
  k_gemm_wmma<true><<<gg, 32, 0, stream>>>(c1, Wp, hW);
  k_init_c<<<gNH, TB, 0, stream>>>(hW, dinv, b2, c2);
  k_edge<<<gEH, TB, 0, stream>>>(src, dst, nrm, hW, c2);

  // layer 3: A = relu(c2)
  k_repack<<<gW, TB, 0, stream>>>(W3, Wp);
  k_gemm_wmma<true><<<gg, 32, 0, stream>>>(c2, Wp, hW);
  k_init_c<<<gNH, TB, 0, stream>>>(hW, dinv, b3, c3);
  k_edge<<<gEH, TB, 0, stream>>>(src, dst, nrm, hW, c3);

  // mean pool over graphs (relu(c3) applied on load)
  k_pool_zero<<<(N_GRAPHS * HDIM + N_GRAPHS + TB - 1) / TB, TB, 0, stream>>>(psum);
  k_pool_acc<<<gNH, TB, 0, stream>>>(c3, batch, psum, cnt);
  k_pool_fin<<<(N_GRAPHS * HDIM + TB - 1) / TB, TB, 0, stream>>>(psum, cnt, pooled);

  // classifier head
  k_mlp<<<1, 128, 0, stream>>>(pooled, L1w, L1b, L2w, L2b, L3w, L3b, out);
}